// Mamba2MCLMHeadModel_80513456930942
// MI455X (gfx1250) — compile-verified
//
#include <hip/hip_runtime.h>
#include <math.h>

// ---- problem constants (from reference) ----
static constexpr int Lc  = 4;
static constexpr int Bc  = 8;
static constexpr int Dc  = 2048;
static constexpr int Dic = 4096;
static constexpr int Nc_ = 128;     // state dim N
static constexpr int Hc  = 64;
static constexpr int Pc  = 64;
static constexpr int CK  = 4;       // conv kernel
static constexpr int Mc  = 8;       // cached chunks
static constexpr int Vc  = 50288;
static constexpr int Cch = Dic + 2 * Nc_;           // 4352
static constexpr int Dout = 2 * Dic + 2 * Nc_ + Hc; // 8512
static constexpr float EPS = 1e-5f;
static constexpr int KCHUNK = 1024;                 // K-tile held in LDS (16 rows)

typedef __attribute__((ext_vector_type(2))) float v2f;
typedef __attribute__((ext_vector_type(8))) float v8f;

// ---------------------------------------------------------------------------
// Embedding gather: res[b][d] = emb[ids[b]][d]
// ---------------------------------------------------------------------------
__global__ void embed_kernel(const int* __restrict__ ids,
                             const float* __restrict__ emb,
                             float* __restrict__ res) {
  int i = blockIdx.x * blockDim.x + threadIdx.x;   // over B*D
  if (i >= Bc * Dc) return;
  int b = i / Dc, d = i % Dc;
  res[i] = emb[(long)ids[b] * Dc + d];
}

// ---------------------------------------------------------------------------
// RMSNorm with optional SiLU gate:  v = in * silu(gate);
// out = v * rsqrt(mean(v^2)+eps) * w.  One block per batch row.
// ---------------------------------------------------------------------------
__global__ void rmsnorm_kernel(const float* __restrict__ in, long inStride,
                               const float* __restrict__ gate, long gateStride,
                               const float* __restrict__ w,
                               float* __restrict__ out, int width) {
  int b = blockIdx.x;
  const float* xi = in + (long)b * inStride;
  const float* gi = gate ? gate + (long)b * gateStride : nullptr;
  __shared__ float red[8];
  float ss = 0.f;
  for (int i = threadIdx.x; i < width; i += blockDim.x) {
    float v = xi[i];
    if (gi) { float g = gi[i]; v *= g / (1.f + __expf(-g)); }
    ss += v * v;
  }
  for (int off = 16; off; off >>= 1) ss += __shfl_xor(ss, off, 32);
  if ((threadIdx.x & 31) == 0) red[threadIdx.x >> 5] = ss;
  __syncthreads();
  if (threadIdx.x == 0) {
    float t = 0.f;
    int nw = blockDim.x >> 5;
    for (int i = 0; i < nw; ++i) t += red[i];
    red[0] = rsqrtf(t / (float)width + EPS);
  }
  __syncthreads();
  float scale = red[0];
  float* yo = out + (long)b * width;
  for (int i = threadIdx.x; i < width; i += blockDim.x) {
    float v = xi[i];
    if (gi) { float g = gi[i]; v *= g / (1.f + __expf(-g)); }
    yo[i] = v * scale * w[i];
  }
}

// ---------------------------------------------------------------------------
// Generic 8-row GEMM via V_WMMA_F32_16X16X4_F32:
//   Out[m][n] (+)= sum_k X[m][k] * W[k*sK + n*sN],  m in [0,8), n in [0,Ncols)
// X is 8 x K row-major (stride K), zero-padded to 16 rows inside LDS so the
// A-operand load is branch-free (EXEC stays all-ones through the WMMA loop).
// B operand streamed from global with generic strides: in_proj/out_proj use
// (sK=Ncols, sN=1); logits uses emb^T in place via (sK=1, sN=D).
// 256 threads = 8 waves; each wave owns one 16-column output tile.
// ---------------------------------------------------------------------------
template <bool ACC>
__global__ __launch_bounds__(256)
void gemm8_wmma(const float* __restrict__ X, const float* __restrict__ W,
                float* __restrict__ Out,
                int K, int Ncols, long sK, long sN) {
  extern __shared__ float xs[];                  // 16 * KC floats (<= 64 KB)
  const int KC = (K < KCHUNK) ? K : KCHUNK;
  const int tid  = threadIdx.x;
  const int lane = tid & 31;
  const int wave = tid >> 5;
  const int hi = lane >> 4;                      // 0: K=0,1 half  1: K=2,3 half
  const int mn = lane & 15;                      // A: row m   B/D: col n
  const long n0 = ((long)blockIdx.x * 8 + wave) * 16;
  const bool active = (n0 < Ncols);

  v8f c = {0.f, 0.f, 0.f, 0.f, 0.f, 0.f, 0.f, 0.f};

  for (int kb = 0; kb < K; kb += KC) {
    // cooperative load of X[0:8][kb:kb+KC]; rows 8..15 zero-filled (padding)
    for (int r = 0; r < 16; ++r)
      for (int kk = tid; kk < KC; kk += blockDim.x)
        xs[r * KC + kk] = (r < 8) ? X[(long)r * K + kb + kk] : 0.f;
    __syncthreads();

    if (active) {
      const float* ap = &xs[mn * KC + 2 * hi];   // A: (m=mn, k=2hi, 2hi+1)
      const float* wb = W + ((long)kb + 2 * hi) * sK + (n0 + mn) * sN;
#pragma unroll 4
      for (int k = 0; k < KC; k += 4) {
        v2f a;
        a.x = ap[k];
        a.y = ap[k + 1];
        const float* wp = wb + (long)k * sK;
        v2f bv;
        bv.x = wp[0];
        bv.y = wp[sK];
        c = __builtin_amdgcn_wmma_f32_16x16x4_f32(false, a, false, bv,
                                                  (short)0, c, false, false);
      }
    }
    __syncthreads();
  }

  // D layout: VGPR r, lanes 0-15 -> M=r (valid rows 0..7), lanes 16-31 -> M=r+8
  if (active && hi == 0) {
#pragma unroll
    for (int r = 0; r < 8; ++r) {
      long idx = (long)r * Ncols + n0 + mn;
      if (ACC) Out[idx] += c[r];
      else     Out[idx]  = c[r];
    }
  }
}

// ---------------------------------------------------------------------------
// Depthwise conv-4 update + SiLU (conv_state is read-only; no writeback in ref)
//   xBC[b][c] = silu( st[c][1]*w0 + st[c][2]*w1 + st[c][3]*w2 + xnew*w3 + bias )
// ---------------------------------------------------------------------------
__global__ void conv_kernel(const float* __restrict__ zx,
                            const float* __restrict__ conv_state,
                            const float* __restrict__ conv_w,
                            const float* __restrict__ conv_b,
                            float* __restrict__ xBC, int l) {
  int c = blockIdx.x * blockDim.x + threadIdx.x;
  int b = blockIdx.y;
  if (c >= Cch) return;
  const float* st = conv_state + ((((long)l * Bc + b) * Cch) + c) * CK;
  const float* w  = conv_w + ((long)l * Cch + c) * CK;
  float xnew = zx[(long)b * Dout + Dic + c];
  float acc = conv_b[(long)l * Cch + c]
            + st[1] * w[0] + st[2] * w[1] + st[3] * w[2] + xnew * w[3];
  acc = acc / (1.f + __expf(-acc));
  xBC[(long)b * Cch + c] = acc;
}

// ---------------------------------------------------------------------------
// dt = softplus(dt_raw + dt_bias);  dA = exp(-exp(A_log)*dt)
// ---------------------------------------------------------------------------
__global__ void dt_kernel(const float* __restrict__ zx,
                          const float* __restrict__ dt_bias,
                          const float* __restrict__ A_log,
                          float* __restrict__ dtw, float* __restrict__ dAw,
                          int l) {
  int i = blockIdx.x * blockDim.x + threadIdx.x;   // over B*H
  if (i >= Bc * Hc) return;
  int b = i / Hc, h = i % Hc;
  float d = zx[(long)b * Dout + Dic + Cch + h] + dt_bias[(long)l * Hc + h];
  float dt = (d > 20.f) ? d : log1pf(__expf(d));
  float dA = __expf(-__expf(A_log[(long)l * Hc + h]) * dt);
  dtw[i] = dt;
  dAw[i] = dA;
}

// ---------------------------------------------------------------------------
// SSM update + multi-chunk blend + C-contraction.  One block per (b,h).
//   fused[p][n] = sum_m w[m]*cached[b][m][h][p][n]
//               + w[M]*(ssm_state[p][n]*dA + dt*Bm[n]*x[p])
//   y[b][h][p]  = sum_n fused[p][n]*Cm[n] + D_param[h]*x[p]
// Thread t: p = t>>2, q = t&3; q-lanes partition n (stride-4, coalesced quads),
// shuffle-reduce over q.
// ---------------------------------------------------------------------------
__global__ __launch_bounds__(256)
void ssm_kernel(const float* __restrict__ xBC,
                const float* __restrict__ dtw, const float* __restrict__ dAw,
                const float* __restrict__ ssm_state,
                const float* __restrict__ cached,
                const float* __restrict__ W_mc,
                const float* __restrict__ D_param,
                float* __restrict__ y, int l) {
  int bh = blockIdx.x;
  int b = bh / Hc, h = bh % Hc;
  __shared__ float wsm[Mc + 1];
  __shared__ float Bs[Nc_], Cs[Nc_];
  int t = threadIdx.x;
  if (t == 0) {
    const float* wm = W_mc + (long)l * (Mc + 1);
    float mx = wm[0];
    for (int i = 1; i <= Mc; ++i) mx = fmaxf(mx, wm[i]);
    float s = 0.f, e[Mc + 1];
    for (int i = 0; i <= Mc; ++i) { e[i] = __expf(wm[i] - mx); s += e[i]; }
    for (int i = 0; i <= Mc; ++i) wsm[i] = e[i] / s;
  }
  if (t < Nc_)            Bs[t]       = xBC[(long)b * Cch + Dic + t];
  else if (t < 2 * Nc_)   Cs[t - Nc_] = xBC[(long)b * Cch + Dic + Nc_ + (t - Nc_)];
  __syncthreads();

  float dt = dtw[bh], dA = dAw[bh], w8 = wsm[Mc];
  int p = t >> 2, q = t & 3;
  float xp = xBC[(long)b * Cch + h * Pc + p];
  const float* sbase = ssm_state + ((((long)l * Bc + b) * Hc + h) * (long)Pc + p) * Nc_;
  const float* cb0   = cached +
      ((((long)l * Bc + b) * Mc * Hc + h) * (long)Pc + p) * Nc_;
  const long strideM = (long)Hc * Pc * Nc_;

  float acc = 0.f;
  for (int i = 0; i < Nc_ / 4; ++i) {
    int n = q + 4 * i;
    float fused = w8 * (sbase[n] * dA + dt * Bs[n] * xp);
#pragma unroll
    for (int m = 0; m < Mc; ++m)
      fused += wsm[m] * cb0[(long)m * strideM + n];
    acc += fused * Cs[n];
  }
  acc += __shfl_xor(acc, 1, 32);
  acc += __shfl_xor(acc, 2, 32);
  if (q == 0)
    y[(long)b * Dic + h * Pc + p] = acc + D_param[(long)l * Hc + h] * xp;
}

// ---------------------------------------------------------------------------
extern "C" void kernel_launch(void* const* d_in, const int* in_sizes, int n_in,
                              void* d_out, int out_size, void* d_ws, size_t ws_size,
                              hipStream_t stream) {
  const int*   ids          = (const int*)  d_in[0];
  const float* emb          = (const float*)d_in[1];
  const float* in_proj_w    = (const float*)d_in[2];
  const float* conv_w       = (const float*)d_in[3];
  const float* conv_b       = (const float*)d_in[4];
  const float* A_log        = (const float*)d_in[5];
  const float* dt_bias      = (const float*)d_in[6];
  const float* D_param      = (const float*)d_in[7];
  const float* mixer_norm_w = (const float*)d_in[8];
  const float* out_proj_w   = (const float*)d_in[9];
  const float* layer_norm_w = (const float*)d_in[10];
  const float* norm_f_w     = (const float*)d_in[11];
  const float* W_mc         = (const float*)d_in[12];
  const float* conv_state   = (const float*)d_in[13];
  const float* ssm_state    = (const float*)d_in[14];
  const float* cached       = (const float*)d_in[15];
  float* logits = (float*)d_out;

  // workspace carve (floats)
  float* ws  = (float*)d_ws;
  float* res = ws;                 ws += Bc * Dc;     // residual
  float* xn  = ws;                 ws += Bc * Dc;     // normed activations
  float* zx  = ws;                 ws += Bc * Dout;   // zxbcdt
  float* xBC = ws;                 ws += Bc * Cch;    // conv output
  float* dtw = ws;                 ws += Bc * Hc;
  float* dAw = ws;                 ws += Bc * Hc;
  float* y   = ws;                 ws += Bc * Dic;
  float* yn  = ws;                 ws += Bc * Dic;

  const size_t lds = 16 * KCHUNK * sizeof(float);    // 64 KB LDS per GEMM block

  embed_kernel<<<(Bc * Dc + 255) / 256, 256, 0, stream>>>(ids, emb, res);

  for (int l = 0; l < Lc; ++l) {
    rmsnorm_kernel<<<Bc, 256, 0, stream>>>(
        res, Dc, nullptr, 0, layer_norm_w + (long)l * Dc, xn, Dc);

    // zxbcdt = xn @ in_proj_w[l]   (8x2048 @ 2048x8512)
    {
      int ntiles = Dout / 16;                        // 532
      int blocks = (ntiles + 7) / 8;
      gemm8_wmma<false><<<blocks, 256, lds, stream>>>(
          xn, in_proj_w + (long)l * Dc * Dout, zx,
          Dc, Dout, (long)Dout, 1L);
    }

    conv_kernel<<<dim3((Cch + 255) / 256, Bc), 256, 0, stream>>>(
        zx, conv_state, conv_w, conv_b, xBC, l);

    dt_kernel<<<(Bc * Hc + 255) / 256, 256, 0, stream>>>(
        zx, dt_bias, A_log, dtw, dAw, l);

    ssm_kernel<<<Bc * Hc, 256, 0, stream>>>(
        xBC, dtw, dAw, ssm_state, cached, W_mc, D_param, y, l);

    // yn = rms(y * silu(z), mixer_norm_w)   (z = zx[:, :Di], row stride Dout)
    rmsnorm_kernel<<<Bc, 256, 0, stream>>>(
        y, Dic, zx, Dout, mixer_norm_w + (long)l * Dic, yn, Dic);

    // res += yn @ out_proj_w[l]   (8x4096 @ 4096x2048), accumulate
    {
      int ntiles = Dc / 16;                          // 128
      int blocks = (ntiles + 7) / 8;
      gemm8_wmma<true><<<blocks, 256, lds, stream>>>(
          yn, out_proj_w + (long)l * Dic * Dc, res,
          Dic, Dc, (long)Dc, 1L);
    }
  }

  rmsnorm_kernel<<<Bc, 256, 0, stream>>>(res, Dc, nullptr, 0, norm_f_w, xn, Dc);

  // logits = xn @ emb^T   (8x2048 @ 2048x50288), W(k,n) = emb[n*D + k]
  {
    int ntiles = Vc / 16;                            // 3143
    int blocks = (ntiles + 7) / 8;
    gemm8_wmma<false><<<blocks, 256, lds, stream>>>(
        xn, emb, logits, Dc, Vc, 1L, (long)Dc);
  }
}